// AdaptiveThresholdCoding_1297080123581
// MI455X (gfx1250) — compile-verified
//
#include <hip/hip_runtime.h>

// Adaptive threshold spike encoding for MI455X (gfx1250, wave32).
//
// Shapes: x[B=32, F=65536] f32 -> spikes[B=32, T=32, F=65536] f32.
// Pure streaming problem: 8 MiB in, 256 MiB out => store-bandwidth bound,
// floor ~11.5 us at 23.3 TB/s. No matmul structure (data-dependent reset in
// the recurrence), so the win is minimal-traffic codegen:
//   - one b128 load per 4 inputs, recurrence state lives in VGPRs
//   - 32 fully-coalesced non-temporal b128 stores per thread (output is
//     write-once and larger than the 192MB L2 -> TH=NT streams past L2)

#define T_STEPS 32
#define B_DIM   32
#define F_DIM   65536
#define F_VEC4  (F_DIM / 4)
#define INIT_THR 0.5f
#define RATE     0.1f

// Native clang vector type: required by __builtin_nontemporal_{load,store},
// and guarantees b128 vector memory ops.
typedef float v4f __attribute__((ext_vector_type(4)));

__global__ __launch_bounds__(256) void
adaptive_threshold_spikes(const v4f* __restrict__ x4, v4f* __restrict__ out4)
{
    const unsigned tid = blockIdx.x * blockDim.x + threadIdx.x;   // over B*F/4
    if (tid >= (unsigned)(B_DIM * F_VEC4)) return;

    const unsigned b  = tid / F_VEC4;
    const unsigned f4 = tid - b * F_VEC4;

    // Single 128-bit load of the 4 input elements this thread owns.
    const v4f xv = x4[tid];

    // Recurrence state in registers.
    float a0 = 0.0f, a1 = 0.0f, a2 = 0.0f, a3 = 0.0f;
    float t0 = INIT_THR, t1 = INIT_THR, t2 = INIT_THR, t3 = INIT_THR;
    const float keep = 1.0f - RATE;                  // 0.9f
    const float d0 = RATE * fabsf(xv.x);
    const float d1 = RATE * fabsf(xv.y);
    const float d2 = RATE * fabsf(xv.z);
    const float d3 = RATE * fabsf(xv.w);

    // Output index for (b, t, f4) in float4 units: ((b*T + t) * F_VEC4) + f4.
    v4f* __restrict__ obase = out4 + ((size_t)b * T_STEPS) * F_VEC4 + f4;

#pragma unroll
    for (int t = 0; t < T_STEPS; ++t) {
        a0 += xv.x; a1 += xv.y; a2 += xv.z; a3 += xv.w;

        const bool m0 = a0 >= t0;
        const bool m1 = a1 >= t1;
        const bool m2 = a2 >= t2;
        const bool m3 = a3 >= t3;

        v4f s;
        s.x = m0 ? 1.0f : 0.0f;
        s.y = m1 ? 1.0f : 0.0f;
        s.z = m2 ? 1.0f : 0.0f;
        s.w = m3 ? 1.0f : 0.0f;

        a0 = m0 ? 0.0f : a0;
        a1 = m1 ? 0.0f : a1;
        a2 = m2 ? 0.0f : a2;
        a3 = m3 ? 0.0f : a3;

        t0 = t0 * keep + d0;
        t1 = t1 * keep + d1;
        t2 = t2 * keep + d2;
        t3 = t3 * keep + d3;

        // Streaming store: write-once output, bypass/deprioritize caches.
        __builtin_nontemporal_store(s, obase + (size_t)t * F_VEC4);
    }
}

extern "C" void kernel_launch(void* const* d_in, const int* in_sizes, int n_in,
                              void* d_out, int out_size, void* d_ws, size_t ws_size,
                              hipStream_t stream)
{
    (void)in_sizes; (void)n_in; (void)out_size; (void)d_ws; (void)ws_size;

    const v4f* x4   = (const v4f*)d_in[0];
    v4f*       out4 = (v4f*)d_out;

    const int total_vec4 = B_DIM * F_VEC4;          // 524288 threads
    const int block = 256;                          // 8 wave32 waves
    const int grid  = (total_vec4 + block - 1) / block;   // 2048 blocks

    adaptive_threshold_spikes<<<grid, block, 0, stream>>>(x4, out4);
}